// GNNPolicy_87625922773436
// MI455X (gfx1250) — compile-verified
//
#include <hip/hip_runtime.h>
#include <hip/hip_bf16.h>

typedef float v2f __attribute__((ext_vector_type(2)));
typedef float v8f __attribute__((ext_vector_type(8)));

#define N_NODES 100000
#define N_EDGES 1600000
#define IN_DIM 64
#define HID 128
#define N_ASSETS 50
#define N_GRAPHS 64

// ---------------------------------------------------------------------------
// Edge scatter-add: agg[dst] += feat[src], thread = (edge, 4-feature chunk).
// Aggregation buffers are <= 51 MB -> resident in the 192 MB L2, so the
// global_atomic_add_f32 traffic resolves on-chip.
// ---------------------------------------------------------------------------
__global__ void __launch_bounds__(256) scatter_add_kernel(
    const float* __restrict__ feat, const int* __restrict__ src,
    const int* __restrict__ dst, float* __restrict__ agg,
    int n_edges, int dim4 /* dim/4 */, int dim) {
  int t = blockIdx.x * 256 + threadIdx.x;
  int total = n_edges * dim4;
  if (t >= total) return;
  int e = t / dim4;
  int f = (t - e * dim4) * 4;
  int s = src[e];
  int d = dst[e];
  float4 v = *(const float4*)(feat + (size_t)s * dim + f);
  float* p = agg + (size_t)d * dim + f;
  atomicAdd(p + 0, v.x);
  atomicAdd(p + 1, v.y);
  atomicAdd(p + 2, v.z);
  atomicAdd(p + 3, v.w);
}

// ---------------------------------------------------------------------------
// Fused GraphConv GEMM:  out = relu(Aagg @ Wrel + Aroot @ Wroot + bias)
// Block = 256 threads = 8 waves. Block owns 16 node rows; both A operands are
// staged in LDS. Wave w owns output columns [16w, 16w+16). K-loop is a fully
// unrolled chain of V_WMMA_F32_16X16X4_F32 (fp32 in, fp32 accum).
//
// f32 WMMA lane layouts (ISA 7.12.2):
//   A 16x4 : lanes 0-15 -> M=lane, VGPR{0,1}=K{0,1}; lanes 16-31 -> K{2,3}
//   B 4x16 : lanes 0-15 -> N=lane, VGPR{0,1}=K{0,1}; lanes 16-31 -> K{2,3}
//   C 16x16: VGPR r -> M=r (lanes 0-15) / M=r+8 (lanes 16-31), N=lane&15
// ---------------------------------------------------------------------------
template <int K>
__global__ void __launch_bounds__(256) gemm_conv_kernel(
    const float* __restrict__ Aagg, const float* __restrict__ Aroot,
    const float* __restrict__ Wrel, const float* __restrict__ Wroot,
    const float* __restrict__ bias, float* __restrict__ out, int n_nodes) {
  __shared__ float sA[2 * 16 * K];

  const int m0 = blockIdx.x * 16;
  const int t = threadIdx.x;

  // Cooperative stage of Aagg tile and Aroot tile (float4 coalesced).
  const int total4 = (2 * 16 * K) / 4;
  for (int idx = t; idx < total4; idx += 256) {
    int flat = idx * 4;
    int which = flat / (16 * K);
    int rem = flat - which * (16 * K);
    int r = rem / K;
    int k = rem - r * K;
    int gr = m0 + r;
    if (gr >= n_nodes) gr = n_nodes - 1;  // clamp: OOB rows never stored
    const float* sp = which ? Aroot : Aagg;
    float4 v = *(const float4*)(sp + (size_t)gr * K + k);
    *(float4*)(&sA[which * 16 * K + r * K + k]) = v;
  }
  __syncthreads();

  const int lane = t & 31;
  const int wave = t >> 5;        // 0..7 -> column tile
  const int row = lane & 15;      // M within tile (A operand)
  const int kh = lane >> 4;       // K-half selector (0 -> K{0,1}, 1 -> K{2,3})
  const int col = wave * 16 + (lane & 15);

  v8f c = {0.f, 0.f, 0.f, 0.f, 0.f, 0.f, 0.f, 0.f};

#pragma unroll
  for (int s = 0; s < 2; ++s) {
    const float* __restrict__ W = s ? Wroot : Wrel;
    const float* As = &sA[s * 16 * K + row * K];
#pragma unroll
    for (int k0 = 0; k0 < K; k0 += 4) {
      const int kk = k0 + 2 * kh;
      v2f a, b;
      a.x = As[kk];
      a.y = As[kk + 1];
      b.x = W[kk * HID + col];
      b.y = W[(kk + 1) * HID + col];
      c = __builtin_amdgcn_wmma_f32_16x16x4_f32(
          /*neg_a=*/false, a, /*neg_b=*/false, b,
          /*c_mod=*/(short)0, c, /*reuse_a=*/false, /*reuse_b=*/false);
    }
  }

  const float bb = bias[col];
#pragma unroll
  for (int r = 0; r < 8; ++r) {
    int gr = m0 + r + 8 * kh;
    if (gr < n_nodes) {
      float v = c[r] + bb;
      out[(size_t)gr * HID + col] = v > 0.f ? v : 0.f;
    }
  }
}

// ---------------------------------------------------------------------------
// Segment-sum pooling: sums[batch[n]] += h[n], cnts[batch[n]] += 1
// ---------------------------------------------------------------------------
__global__ void __launch_bounds__(256) pool_sum_kernel(
    const float* __restrict__ h, const int* __restrict__ batch,
    float* __restrict__ sums, float* __restrict__ cnts, int n_nodes) {
  int t = blockIdx.x * 256 + threadIdx.x;
  int total = n_nodes * (HID / 4);
  if (t >= total) return;
  int n = t / (HID / 4);
  int f = (t - n * (HID / 4)) * 4;
  int g = batch[n];
  float4 v = *(const float4*)(h + (size_t)n * HID + f);
  float* p = sums + (size_t)g * HID + f;
  atomicAdd(p + 0, v.x);
  atomicAdd(p + 1, v.y);
  atomicAdd(p + 2, v.z);
  atomicAdd(p + 3, v.w);
  if (f == 0) atomicAdd(&cnts[g], 1.0f);
}

// ---------------------------------------------------------------------------
// MLP head: pooled mean -> fc1+relu -> fc2 logits (single block, tiny work)
// ---------------------------------------------------------------------------
__global__ void __launch_bounds__(256) head_fc_kernel(
    const float* __restrict__ sums, const float* __restrict__ cnts,
    const float* __restrict__ fc1w, const float* __restrict__ fc1b,
    const float* __restrict__ fc2w, const float* __restrict__ fc2b,
    float* __restrict__ logits) {
  __shared__ float pm[N_GRAPHS * HID];
  __shared__ float o1[N_GRAPHS * HID];
  const int t = threadIdx.x;

  for (int i = t; i < N_GRAPHS * HID; i += 256) {
    int g = i >> 7;
    float cv = cnts[g];
    cv = cv > 1.f ? cv : 1.f;
    pm[i] = sums[i] / cv;
  }
  __syncthreads();

  for (int i = t; i < N_GRAPHS * HID; i += 256) {
    int g = i >> 7, j = i & (HID - 1);
    float acc = fc1b[j];
    for (int k = 0; k < HID; ++k) acc = fmaf(pm[g * HID + k], fc1w[k * HID + j], acc);
    o1[i] = acc > 0.f ? acc : 0.f;
  }
  __syncthreads();

  for (int i = t; i < N_GRAPHS * N_ASSETS; i += 256) {
    int g = i / N_ASSETS, j = i - g * N_ASSETS;
    float acc = fc2b[j];
    for (int k = 0; k < HID; ++k) acc = fmaf(o1[g * HID + k], fc2w[k * N_ASSETS + j], acc);
    logits[g * N_ASSETS + j] = acc;
  }
}

__global__ void __launch_bounds__(64) softmax_kernel(
    const float* __restrict__ logits, float* __restrict__ out) {
  int g = threadIdx.x;
  if (g >= N_GRAPHS) return;
  const float* l = logits + g * N_ASSETS;
  float m = -1e30f;
  for (int j = 0; j < N_ASSETS; ++j) m = fmaxf(m, l[j]);
  float s = 0.f;
  for (int j = 0; j < N_ASSETS; ++j) s += expf(l[j] - m);
  float inv = 1.f / s;
  for (int j = 0; j < N_ASSETS; ++j) out[g * N_ASSETS + j] = expf(l[j] - m) * inv;
}

// ---------------------------------------------------------------------------
extern "C" void kernel_launch(void* const* d_in, const int* in_sizes, int n_in,
                              void* d_out, int out_size, void* d_ws, size_t ws_size,
                              hipStream_t stream) {
  const float* x       = (const float*)d_in[0];
  const int*   ei      = (const int*)d_in[1];
  const int*   batch   = (const int*)d_in[2];
  const float* w1_rel  = (const float*)d_in[3];
  const float* b1      = (const float*)d_in[4];
  const float* w1_root = (const float*)d_in[5];
  const float* w2_rel  = (const float*)d_in[6];
  const float* b2      = (const float*)d_in[7];
  const float* w2_root = (const float*)d_in[8];
  const float* fc1w    = (const float*)d_in[9];
  const float* fc1b    = (const float*)d_in[10];
  const float* fc2w    = (const float*)d_in[11];
  const float* fc2b    = (const float*)d_in[12];
  float* out = (float*)d_out;

  const int* src = ei;
  const int* dst = ei + N_EDGES;

  // Workspace carve-up (256B aligned)
  char* ws = (char*)d_ws;
  size_t off = 0;
  auto carve = [&](size_t bytes) {
    char* p = ws + off;
    off += (bytes + 255) & ~size_t(255);
    return p;
  };
  float* agg1   = (float*)carve((size_t)N_NODES * IN_DIM * 4);  // 25.6 MB
  float* h1     = (float*)carve((size_t)N_NODES * HID * 4);     // 51.2 MB
  float* agg2   = (float*)carve((size_t)N_NODES * HID * 4);     // 51.2 MB
  float* h2     = (float*)carve((size_t)N_NODES * HID * 4);     // 51.2 MB
  float* psums  = (float*)carve((size_t)N_GRAPHS * HID * 4);
  float* pcnts  = (float*)carve((size_t)N_GRAPHS * 4);
  float* logits = (float*)carve((size_t)N_GRAPHS * N_ASSETS * 4);

  // Zero accumulation buffers every call (deterministic, capture-legal)
  hipMemsetAsync(agg1, 0, (size_t)N_NODES * IN_DIM * 4, stream);
  hipMemsetAsync(agg2, 0, (size_t)N_NODES * HID * 4, stream);
  hipMemsetAsync(psums, 0, (size_t)N_GRAPHS * HID * 4, stream);
  hipMemsetAsync(pcnts, 0, (size_t)N_GRAPHS * 4, stream);

  const int mtiles = (N_NODES + 15) / 16;

  // Layer 1: aggregate raw x, then fused dual-GEMM + bias + relu
  {
    int total = N_EDGES * (IN_DIM / 4);
    scatter_add_kernel<<<(total + 255) / 256, 256, 0, stream>>>(
        x, src, dst, agg1, N_EDGES, IN_DIM / 4, IN_DIM);
    gemm_conv_kernel<IN_DIM><<<mtiles, 256, 0, stream>>>(
        agg1, x, w1_rel, w1_root, b1, h1, N_NODES);
  }

  // Layer 2
  {
    int total = N_EDGES * (HID / 4);
    scatter_add_kernel<<<(total + 255) / 256, 256, 0, stream>>>(
        h1, src, dst, agg2, N_EDGES, HID / 4, HID);
    gemm_conv_kernel<HID><<<mtiles, 256, 0, stream>>>(
        agg2, h1, w2_rel, w2_root, b2, h2, N_NODES);
  }

  // Mean pool + head
  {
    int total = N_NODES * (HID / 4);
    pool_sum_kernel<<<(total + 255) / 256, 256, 0, stream>>>(
        h2, batch, psums, pcnts, N_NODES);
    head_fc_kernel<<<1, 256, 0, stream>>>(psums, pcnts, fc1w, fc1b, fc2w, fc2b, logits);
    softmax_kernel<<<1, 64, 0, stream>>>(logits, out);
  }
}